// TSDFFusion_80135499809391
// MI455X (gfx1250) — compile-verified
//
#include <hip/hip_runtime.h>

#define GRID_N (256 * 256 * 256)
#define IMG_H 720
#define IMG_W 1280
#define TRUNC_INV 10.0f
#define MAX_WEIGHT 100.0f

typedef float v2f __attribute__((ext_vector_type(2)));
typedef float v8f __attribute__((ext_vector_type(8)));

// ---------------------------------------------------------------------------
// Prep (1 thread): world2cam = inv(pose) via Gauss-Jordan w/ partial pivoting
// (matches jnp.linalg.inv for general 4x4), then fold intrinsics:
//   M[i][:] = sum_k K[i][k] * W2C[k][:]  (i = 0..2, acting on [x,y,z,1])
//   M[3][:] = W2C[2][:]                  (camera-space depth row)
// Store 16 floats row-major into d_ws.
// ---------------------------------------------------------------------------
__global__ void tsdf_prep(const float* __restrict__ pose,
                          const float* __restrict__ K,
                          float* __restrict__ M4out) {
  if (threadIdx.x != 0 || blockIdx.x != 0) return;
  float a[4][8];
  for (int i = 0; i < 4; ++i)
    for (int j = 0; j < 4; ++j) {
      a[i][j] = pose[i * 4 + j];
      a[i][4 + j] = (i == j) ? 1.0f : 0.0f;
    }
  for (int col = 0; col < 4; ++col) {
    int piv = col;
    float best = fabsf(a[col][col]);
    for (int r = col + 1; r < 4; ++r) {
      float v = fabsf(a[r][col]);
      if (v > best) { best = v; piv = r; }
    }
    if (piv != col)
      for (int j = 0; j < 8; ++j) {
        float t = a[col][j]; a[col][j] = a[piv][j]; a[piv][j] = t;
      }
    float inv = 1.0f / a[col][col];
    for (int j = 0; j < 8; ++j) a[col][j] *= inv;
    for (int r = 0; r < 4; ++r)
      if (r != col) {
        float f = a[r][col];
        for (int j = 0; j < 8; ++j) a[r][j] -= f * a[col][j];
      }
  }
  float W2C[4][4];
  for (int i = 0; i < 4; ++i)
    for (int j = 0; j < 4; ++j) W2C[i][j] = a[i][4 + j];

  float M[4][4];
  for (int i = 0; i < 3; ++i)
    for (int j = 0; j < 4; ++j) {
      float s = 0.0f;
      for (int k = 0; k < 3; ++k) s += K[i * 3 + k] * W2C[k][j];
      M[i][j] = s;
    }
  for (int j = 0; j < 4; ++j) M[3][j] = W2C[2][j];

  for (int i = 0; i < 4; ++i)
    for (int j = 0; j < 4; ++j) M4out[i * 4 + j] = M[i][j];
}

// ---------------------------------------------------------------------------
// Main fuse kernel: 1 voxel/lane, projection via V_WMMA_F32_16X16X4_F32.
//
// WMMA shapes: D(16x16) = A(16x4) x B(4x16).
// A layout (ISA 7.12.2, 32-bit A 16x4):  v0: lane m -> A[m][0], lane 16+m -> A[m][2]
//                                        v1: lane m -> A[m][1], lane 16+m -> A[m][3]
// B layout (K split mirrors A):          v0: lane n -> B[0][n], lane 16+n -> B[2][n]
//                                        v1: lane n -> B[1][n], lane 16+n -> B[3][n]
// D layout (documented):                 vj: lanes 0-15 -> (M=j, N=lane),
//                                            lanes 16-31 -> (M=8+j, N=lane-16)
//
// WMMA#1: A rows 0..3  = M4  -> lanes 0..15 get their voxel's M4*[x,y,z,1] in D[0..3]
// WMMA#2: A rows 8..11 = M4  -> lanes 16..31 get theirs in D[0..3]
//
// Cache policy (CDNA5 TH, ISA §7.3): all one-touch streams (coords, tsdf,
// weight, both outputs) use non-temporal hints so the heavily-reused 3.7 MB
// depth map stays resident in WGP$/L2 while ~470 MB streams past at HBM rate.
// ---------------------------------------------------------------------------
__global__ __launch_bounds__(256) void tsdf_fuse(
    const float* __restrict__ depth_map,
    const float* __restrict__ tsdf_in,
    const float* __restrict__ wgt_in,
    const float* __restrict__ coords,
    const float* __restrict__ M4,
    float* __restrict__ tsdf_out,
    float* __restrict__ wgt_out) {
  const int gid = blockIdx.x * blockDim.x + threadIdx.x;
  const int lane = threadIdx.x & 31;
  const bool lo = lane < 16;
  const int rlo = lane & 15;

  // per-lane voxel coordinates (coalesced 12B-stride reads, non-temporal)
  const float x = __builtin_nontemporal_load(coords + 3 * gid + 0);
  const float y = __builtin_nontemporal_load(coords + 3 * gid + 1);
  const float z = __builtin_nontemporal_load(coords + 3 * gid + 2);

  // A operands: rows 0..3 (WMMA#1) / rows 8..11 (WMMA#2) carry M4, rest zero
  const int c0 = lo ? 0 : 2;
  const int c1 = lo ? 1 : 3;
  float a1v0 = (rlo < 4) ? M4[rlo * 4 + c0] : 0.0f;
  float a1v1 = (rlo < 4) ? M4[rlo * 4 + c1] : 0.0f;
  const int r2 = rlo - 8;
  float a2v0 = (r2 >= 0 && r2 < 4) ? M4[r2 * 4 + c0] : 0.0f;
  float a2v1 = (r2 >= 0 && r2 < 4) ? M4[r2 * 4 + c1] : 0.0f;

  // B operands: column n = [x,y,z,1] of voxel n. One xor-16 shuffle/coord.
  const float xs = __shfl(x, lane ^ 16, 32);
  const float ys = __shfl(y, lane ^ 16, 32);
  const float zs = __shfl(z, lane ^ 16, 32);
  float b1v0 = lo ? x : zs;    // K0 = x_n          | K2 = z_n
  float b1v1 = lo ? y : 1.0f;  // K1 = y_n          | K3 = 1
  float b2v0 = lo ? xs : z;    // K0 = x_{n+16}     | K2 = z_{n+16}
  float b2v1 = lo ? ys : 1.0f; // K1 = y_{n+16}     | K3 = 1

  v2f a1 = {a1v0, a1v1}, a2 = {a2v0, a2v1};
  v2f b1 = {b1v0, b1v1}, b2 = {b2v0, b2v1};
  v8f cz = {};

  v8f d1 = __builtin_amdgcn_wmma_f32_16x16x4_f32(false, a1, false, b1,
                                                 (short)0, cz, false, false);
  v8f d2 = __builtin_amdgcn_wmma_f32_16x16x4_f32(false, a2, false, b2,
                                                 (short)0, cz, false, false);

  const float px = lo ? d1[0] : d2[0];
  const float py = lo ? d1[1] : d2[1];
  const float pz = lo ? d1[2] : d2[2];
  const float vdepth = lo ? d1[3] : d2[3];

  // uv = pix.xy / (pix.z + 1e-8), validity, nearest-pixel gather
  const float invz = 1.0f / (pz + 1e-8f);
  const float u = px * invz;
  const float v = py * invz;
  const bool valid_proj =
      (u >= 0.0f) && (u < (float)IMG_W) && (v >= 0.0f) && (v < (float)IMG_H) &&
      (vdepth > 0.0f);
  int ui = (int)rintf(u);  // round-half-even matches jnp.round
  int vi = (int)rintf(v);
  ui = min(max(ui, 0), IMG_W - 1);
  vi = min(max(vi, 0), IMG_H - 1);
  // reused gather target: default (RT) temporal hint, stays cached
  const float obs = depth_map[vi * IMG_W + ui];

  const float sd = obs - vdepth;
  float tv = sd * TRUNC_INV;
  tv = fminf(fmaxf(tv, -1.0f), 1.0f);
  const bool valid = valid_proj && (obs > 0.0f);
  const float fw = valid ? 1.0f : 0.0f;

  const float t = __builtin_nontemporal_load(tsdf_in + gid);
  const float w = __builtin_nontemporal_load(wgt_in + gid);
  const float nw = w + fw;
  const float out_t = (nw > 0.0f) ? (w * t + fw * tv) / nw : t;
  const float out_w = fminf(nw, MAX_WEIGHT);

  __builtin_nontemporal_store(out_t, tsdf_out + gid);
  __builtin_nontemporal_store(out_w, wgt_out + gid);
}

// ---------------------------------------------------------------------------
extern "C" void kernel_launch(void* const* d_in, const int* in_sizes, int n_in,
                              void* d_out, int out_size, void* d_ws,
                              size_t ws_size, hipStream_t stream) {
  const float* depth = (const float*)d_in[0];   // (720,1280)
  const float* pose = (const float*)d_in[1];    // (4,4)
  const float* K = (const float*)d_in[2];       // (3,3)
  const float* tsdf = (const float*)d_in[3];    // (256,256,256)
  const float* wgt = (const float*)d_in[4];     // (256,256,256)
  const float* coords = (const float*)d_in[5];  // (256^3, 3)

  float* out_tsdf = (float*)d_out;
  float* out_wgt = out_tsdf + GRID_N;
  float* M4 = (float*)d_ws;  // 16 floats

  tsdf_prep<<<1, 1, 0, stream>>>(pose, K, M4);
  tsdf_fuse<<<GRID_N / 256, 256, 0, stream>>>(depth, tsdf, wgt, coords, M4,
                                              out_tsdf, out_wgt);
}